// ModelStage_33019708571697
// MI455X (gfx1250) — compile-verified
//
#include <hip/hip_runtime.h>
#include <math.h>

typedef __attribute__((ext_vector_type(2))) float v2f;
typedef __attribute__((ext_vector_type(8))) float v8f;

#define H 256
#define W 256
#define NPIX 65536      // 256*256
#define CH 24
#define PN 63           // penalty_num
#define KTOT 600        // 25 taps * 24 ch   (tap-major: k = tap*24 + ci)

// ---------------------------------------------------------------------------
// K0: 1/||f0||_F and 1/||f1||_F  (single block)
// ---------------------------------------------------------------------------
__global__ void norms_kernel(const float* __restrict__ f0,
                             const float* __restrict__ f1,
                             float* __restrict__ norms) {
    __shared__ float red[256];
    int tid = threadIdx.x;
    float s0 = 0.f, s1 = 0.f;
    for (int i = tid; i < 600; i += 256)   { float v = f0[i]; s0 += v * v; }
    for (int i = tid; i < 14400; i += 256) { float v = f1[i]; s1 += v * v; }
    red[tid] = s0; __syncthreads();
    for (int s = 128; s > 0; s >>= 1) {
        if (tid < s) red[tid] += red[tid + s];
        __syncthreads();
    }
    if (tid == 0) norms[0] = 1.0f / sqrtf(red[0]);
    __syncthreads();
    red[tid] = s1; __syncthreads();
    for (int s = 128; s > 0; s >>= 1) {
        if (tid < s) red[tid] += red[tid + s];
        __syncthreads();
    }
    if (tid == 0) norms[1] = 1.0f / sqrtf(red[0]);
}

// ---------------------------------------------------------------------------
// K1: repack f1 into paired WMMA A layout.
//   k = tap*24 + ci  (tap = kh*5+kw), m = out-channel (24..31 zero padded)
//   storage: float2 pairs  apack[(k/2)*32 + m] = { A[m][k], A[m][k+1] }
//   forward:   A[m][k] = f1[m][ci][kh][kw] / ||f1||
//   transpose: A[o][k] = f1[ci][o][4-kh][4-kw] / ||f1||   (convT as conv)
// ---------------------------------------------------------------------------
__global__ void pack_f1_kernel(const float* __restrict__ f1,
                               const float* __restrict__ norms,
                               float* __restrict__ apack,
                               float* __restrict__ apackT) {
    int idx = blockIdx.x * blockDim.x + threadIdx.x;   // 600*32
    if (idx >= KTOT * 32) return;
    int k = idx >> 5;
    int m = idx & 31;
    int tap = k / 24;
    int ci  = k - tap * 24;
    int kh  = tap / 5;
    int kw  = tap - kh * 5;
    float inv = norms[1];
    float a  = 0.f, at = 0.f;
    if (m < CH) {
        a  = f1[m * 600 + ci * 25 + tap] * inv;
        at = f1[ci * 600 + m * 25 + (4 - kh) * 5 + (4 - kw)] * inv;
    }
    int dst = (((k >> 1) * 32 + m) << 1) + (k & 1);    // float2-paired layout
    apack[dst]  = a;
    apackT[dst] = at;
}

// ---------------------------------------------------------------------------
// RBF helpers (63-component Gaussian mixture)
// ---------------------------------------------------------------------------
template <bool DERIV>
__device__ __forceinline__ float rbf_eval(float v, const float* __restrict__ sm,
                                          const float* __restrict__ sw, int ch) {
    float s = 0.f;
    const float* wrow = sw + ch * PN;
    for (int j = 0; j < PN; ++j) {
        float d = v - sm[j];
        float e = __expf(d * d * (-1.0f / 200.0f));
        if (DERIV) s += e * d * (-1.0f / 100.0f) * wrow[j];
        else       s += e * wrow[j];
    }
    return s;
}

// ---------------------------------------------------------------------------
// K2: conv 1->24 (edge-replicate pad via clamp) fused with RBF(actw0)
// ---------------------------------------------------------------------------
__global__ void conv1_rbf_kernel(const float* __restrict__ x,
                                 const float* __restrict__ f0,
                                 const float* __restrict__ norms,
                                 const float* __restrict__ mean,
                                 const float* __restrict__ actw,
                                 float* __restrict__ out) {
    __shared__ float sm[PN];
    __shared__ float sw[CH * PN];
    for (int i = threadIdx.x; i < PN; i += blockDim.x)      sm[i] = mean[i];
    for (int i = threadIdx.x; i < CH * PN; i += blockDim.x) sw[i] = actw[i];
    __syncthreads();

    int idx = blockIdx.x * blockDim.x + threadIdx.x;       // 24*65536
    if (idx >= CH * NPIX) return;
    int c  = idx >> 16;
    int p  = idx & (NPIX - 1);
    int yy = p >> 8;
    int xx = p & 255;
    float inv = norms[0];
    float acc = 0.f;
#pragma unroll
    for (int kh = 0; kh < 5; ++kh) {
        int ay = min(max(yy + kh - 2, 0), H - 1);
#pragma unroll
        for (int kw = 0; kw < 5; ++kw) {
            int ax = min(max(xx + kw - 2, 0), W - 1);
            acc += f0[c * 25 + kh * 5 + kw] * x[(ay << 8) + ax];
        }
    }
    acc *= inv;
    out[idx] = rbf_eval<false>(acc, sm, sw, c);
}

// ---------------------------------------------------------------------------
// K3/K4: 24->24 5x5 conv as implicit GEMM on V_WMMA_F32_16X16X4_F32,
// fused RBF / RBF-derivative epilogue.
//   M = out-channel (two 16-tiles; 24..31 zero in apack)
//   N = 16 contiguous pixels of one image row
//   K = 600 tap-major, step 4 (all 4 share one tap since 24 % 4 == 0)
// One wave per 32x16 tile; 4 waves/block; 4096 tiles.
// A frag (fp32 16x4 layout): lanes 0-15 -> {K,K+1}, lanes 16-31 -> {K+2,K+3}.
// ---------------------------------------------------------------------------
template <bool CLAMP, bool DERIV>
__global__ __launch_bounds__(128)
void conv24_wmma_rbf_kernel(const float* __restrict__ in,     // [24,256,256]
                            const float* __restrict__ apack,  // paired [300][32]f2
                            const float* __restrict__ mean,   // [63]
                            const float* __restrict__ actw,   // [24][63]
                            float* __restrict__ out) {        // [24,256,256]
    __shared__ float sm[PN];
    __shared__ float sw[CH * PN];
    for (int i = threadIdx.x; i < PN; i += blockDim.x)      sm[i] = mean[i];
    for (int i = threadIdx.x; i < CH * PN; i += blockDim.x) sw[i] = actw[i];
    __syncthreads();

    const int lane = threadIdx.x & 31;
    const int wave = threadIdx.x >> 5;
    const int tile = blockIdx.x * 4 + wave;      // [0, 4096)
    const int y0   = tile >> 4;                  // image row
    const int x0   = (tile & 15) << 4;           // first pixel of tile
    const int n    = lane & 15;
    const int xn   = x0 + n;
    const bool hi  = lane >= 16;
    const int m    = lane & 15;
    const int koff = hi ? 2 : 0;                 // K sub-offset for this half-wave
    const int k2off = hi ? 1 : 0;                // pair index sub-offset

    v8f c0 = {};
    v8f c1 = {};

    for (int tap = 0; tap < 25; ++tap) {
        const int kh = tap / 5;
        const int kw = tap - kh * 5;
        int yy = y0 + kh - 2;
        int xx = xn + kw - 2;
        bool ok = true;
        if (!CLAMP) ok = (yy >= 0) & (yy < H) & (xx >= 0) & (xx < W);
        yy = min(max(yy, 0), H - 1);
        xx = min(max(xx, 0), W - 1);
        const float* bptr = in + ((yy << 8) + xx);   // + ci*NPIX per channel

#pragma unroll
        for (int ci = 0; ci < CH; ci += 4) {
            const int k2 = tap * 12 + (ci >> 1) + k2off;   // pair index
            v2f aLo = *(const v2f*)(apack + (((k2 * 32) + m)      << 1));
            v2f aHi = *(const v2f*)(apack + (((k2 * 32) + m + 16) << 1));
            const int ca = ci + koff;
            v2f b;
            b.x = ok ? bptr[(ca)     << 16] : 0.0f;
            b.y = ok ? bptr[(ca + 1) << 16] : 0.0f;
            c0 = __builtin_amdgcn_wmma_f32_16x16x4_f32(false, aLo, false, b,
                                                       (short)0, c0, false, false);
            c1 = __builtin_amdgcn_wmma_f32_16x16x4_f32(false, aHi, false, b,
                                                       (short)0, c1, false, false);
        }
    }

    // D-tile: VGPR r -> M = r (lanes 0-15) / r+8 (lanes 16-31); N = lane%16.
    const int mbase = hi ? 8 : 0;
#pragma unroll
    for (int r = 0; r < 8; ++r) {
        int ch = mbase + r;                       // channels 0..15 (tile 0)
        float s = rbf_eval<DERIV>(c0[r], sm, sw, ch);
        out[(ch << 16) + (y0 << 8) + xn] = s;
    }
    if (!hi) {
#pragma unroll
        for (int r = 0; r < 8; ++r) {
            int ch = 16 + r;                      // channels 16..23 (tile 1)
            float s = rbf_eval<DERIV>(c1[r], sm, sw, ch);
            out[(ch << 16) + (y0 << 8) + xn] = s;
        }
    }
}

// ---------------------------------------------------------------------------
// K5: convT with f0 (== 24->1 conv, zero pad, flipped kernel) + final combine
//     out = x - (t + exp(lam_param) * (x - y))
// ---------------------------------------------------------------------------
__global__ void convT0_final_kernel(const float* __restrict__ t,   // [24,256,256]
                                    const float* __restrict__ x,
                                    const float* __restrict__ y,
                                    const float* __restrict__ f0,
                                    const float* __restrict__ norms,
                                    const float* __restrict__ lam_param,
                                    float* __restrict__ out) {
    int p = blockIdx.x * blockDim.x + threadIdx.x;   // 65536
    if (p >= NPIX) return;
    int yy = p >> 8;
    int xx = p & 255;
    float inv = norms[0];
    float acc = 0.f;
    for (int c = 0; c < CH; ++c) {
#pragma unroll
        for (int kh = 0; kh < 5; ++kh) {
            int ay = yy + kh - 2;
            if (ay < 0 || ay >= H) continue;
#pragma unroll
            for (int kw = 0; kw < 5; ++kw) {
                int ax = xx + kw - 2;
                if (ax < 0 || ax >= W) continue;
                acc += t[(c << 16) + (ay << 8) + ax]
                     * f0[c * 25 + (4 - kh) * 5 + (4 - kw)];
            }
        }
    }
    acc *= inv;
    float el = __expf(lam_param[0]);
    float xv = x[p], yv = y[p];
    out[p] = xv - (acc + el * (xv - yv));
}

// ---------------------------------------------------------------------------
// Launch: inputs order = x, y, lam, filter0, filter1, actw0, actw1, mean, lam_param
// ---------------------------------------------------------------------------
extern "C" void kernel_launch(void* const* d_in, const int* in_sizes, int n_in,
                              void* d_out, int out_size, void* d_ws, size_t ws_size,
                              hipStream_t stream) {
    const float* x       = (const float*)d_in[0];
    const float* y       = (const float*)d_in[1];
    const float* f0      = (const float*)d_in[3];
    const float* f1      = (const float*)d_in[4];
    const float* actw0   = (const float*)d_in[5];
    const float* actw1   = (const float*)d_in[6];
    const float* mean    = (const float*)d_in[7];
    const float* lam_par = (const float*)d_in[8];
    float* out = (float*)d_out;

    float* w       = (float*)d_ws;
    float* norms   = w;                            // 2 floats (+pad)
    float* apack   = w + 16;                       // 600*32
    float* apackT  = apack + KTOT * 32;            // 600*32
    float* bufA    = w + 38432;                    // 24*256*256 (128B aligned)
    float* bufB    = bufA + CH * NPIX;             // 24*256*256

    // K0: filter norms
    norms_kernel<<<1, 256, 0, stream>>>(f0, f1, norms);
    // K1: pack f1 (forward + transposed/flipped) into paired WMMA A layout
    pack_f1_kernel<<<(KTOT * 32 + 255) / 256, 256, 0, stream>>>(f1, norms, apack, apackT);
    // K2: conv 1->24 (edge pad) + RBF(actw0) -> bufA
    conv1_rbf_kernel<<<(CH * NPIX) / 256, 256, 0, stream>>>(x, f0, norms, mean, actw0, bufA);
    // K3: conv 24->24 (edge pad) + RBF(actw1) -> bufB      [WMMA]
    conv24_wmma_rbf_kernel<true, false><<<1024, 128, 0, stream>>>(bufA, apack, mean, actw1, bufB);
    // K4: convT(f1) as conv 24->24 (zero pad) + RBF'(actw0) -> bufA   [WMMA]
    conv24_wmma_rbf_kernel<false, true><<<1024, 128, 0, stream>>>(bufB, apackT, mean, actw0, bufA);
    // K5: convT(f0) 24->1 (zero pad) + final combine -> out
    convT0_final_kernel<<<NPIX / 256, 256, 0, stream>>>(bufA, x, y, f0, norms, lam_par, out);
}